// TransHScore_76124000354695
// MI455X (gfx1250) — compile-verified
//
#include <hip/hip_runtime.h>

#define D        128
#define WAVE     32
#define WPB      8            // waves per block
#define BLOCK    (WPB * WAVE) // 256 threads = 8 wave32
#define GAMMA_F  12.0f
#define EPS_F    1e-12f

// ---- gfx1250 async load-to-LDS path (ASYNCcnt), guarded so compile never fails
#if defined(__has_builtin)
#if __has_builtin(__builtin_amdgcn_global_load_async_to_lds_b128) && \
    __has_builtin(__builtin_amdgcn_s_wait_asynccnt)
#define USE_ASYNC_LDS 1
#endif
#endif
#ifndef USE_ASYNC_LDS
#define USE_ASYNC_LDS 0
#endif

// Builtin signature (probed via hipcc diagnostic, round 1):
//   param 1: 'int __vector(4) __device__ *'  -> v4i in addrspace(1), non-const
typedef int v4i __attribute__((ext_vector_type(4)));
typedef __attribute__((address_space(1))) v4i* gv4i_p;   // global (AS1)
typedef __attribute__((address_space(3))) v4i* lv4i_p;   // LDS (AS3)

// DS_SWIZZLE_B32 masked-swizzle patterns: offset[15]=0, [14:10]=xor, [9:5]=or, [4:0]=and
// SWAPXk = (k << 10) | 0x1f  -- butterfly exchange across full wave32.
__device__ __forceinline__ float wave_sum32(float v) {
    v += __int_as_float(__builtin_amdgcn_ds_swizzle(__float_as_int(v), (16 << 10) | 0x1f));
    v += __int_as_float(__builtin_amdgcn_ds_swizzle(__float_as_int(v), (8  << 10) | 0x1f));
    v += __int_as_float(__builtin_amdgcn_ds_swizzle(__float_as_int(v), (4  << 10) | 0x1f));
    v += __int_as_float(__builtin_amdgcn_ds_swizzle(__float_as_int(v), (2  << 10) | 0x1f));
    v += __int_as_float(__builtin_amdgcn_ds_swizzle(__float_as_int(v), (1  << 10) | 0x1f));
    return v;
}

// Hot kernel first (so the disasm snippet shows it): per-edge TransH score.
//   u = head - tail; c = u.n;  score = GAMMA - sum|u - c*n + rel|
// One wave per edge, float4 per lane, grid-stride with 1-iteration-ahead
// index load + row prefetch (global_prefetch_b8). rel row is streamed to LDS
// via the gfx1250 async path (ASYNCcnt) -- it is only consumed after the
// dot-product reduction, so its latency is fully hidden.
__global__ __launch_bounds__(BLOCK)
void transh_score_kernel(const float* __restrict__ node,
                         const float* __restrict__ nrm,   // (pre)normalized normals
                         const float* __restrict__ rel,
                         const int* __restrict__ src,
                         const int* __restrict__ dst,
                         const int* __restrict__ rid,
                         float* __restrict__ out,
                         int E, int waves_total, int prenorm) {
#if USE_ASYNC_LDS
    __shared__ float lds_rel[WPB * D];
    float* slot = lds_rel + (threadIdx.x >> 5) * D;
#endif
    const int lane = threadIdx.x & (WAVE - 1);
    int e = blockIdx.x * WPB + (threadIdx.x >> 5);
    if (e >= E) return;

    int s = src[e], d = dst[e], r = rid[e];

    while (true) {
        const int e2 = e + waves_total;
        int s2 = s, d2 = d, r2 = r;
        if (e2 < E) {
            // Pull next iteration's indices early and prefetch its rows:
            // 32 lanes x 16B strides cover each full 512B row.
            s2 = src[e2]; d2 = dst[e2]; r2 = rid[e2];
            __builtin_prefetch(node + (size_t)s2 * D + lane * 4, 0, 3);
            __builtin_prefetch(node + (size_t)d2 * D + lane * 4, 0, 3);
            __builtin_prefetch(nrm  + (size_t)r2 * D + lane * 4, 0, 3);
            __builtin_prefetch(rel  + (size_t)r2 * D + lane * 4, 0, 3);
        }

        const float4* H  = reinterpret_cast<const float4*>(node + (size_t)s * D);
        const float4* T  = reinterpret_cast<const float4*>(node + (size_t)d * D);
        const float4* Nv = reinterpret_cast<const float4*>(nrm  + (size_t)r * D);
        const float4* Rv = reinterpret_cast<const float4*>(rel  + (size_t)r * D);

#if USE_ASYNC_LDS
        // Stage rel row into this wave's private LDS slot while we do the dot.
        __builtin_amdgcn_global_load_async_to_lds_b128(
            (gv4i_p)(Rv + lane),
            (lv4i_p)(slot + lane * 4), 0, 0);
#endif
        float4 h = H[lane];
        float4 t = T[lane];
        float4 n = Nv[lane];

        if (!prenorm) {  // fallback: normalize the hyperplane normal inline
            float ss = wave_sum32(n.x * n.x + n.y * n.y + n.z * n.z + n.w * n.w);
            float inv = 1.0f / fmaxf(sqrtf(ss), EPS_F);
            n.x *= inv; n.y *= inv; n.z *= inv; n.w *= inv;
        }

        float4 u = make_float4(h.x - t.x, h.y - t.y, h.z - t.z, h.w - t.w);
        float c = wave_sum32(u.x * n.x + u.y * n.y + u.z * n.z + u.w * n.w);

#if USE_ASYNC_LDS
        __builtin_amdgcn_s_wait_asynccnt(0);
        float4 rv = reinterpret_cast<const float4*>(slot)[lane];
#else
        float4 rv = Rv[lane];
#endif
        float ax = fabsf(fmaf(-c, n.x, u.x) + rv.x);
        float ay = fabsf(fmaf(-c, n.y, u.y) + rv.y);
        float az = fabsf(fmaf(-c, n.z, u.z) + rv.z);
        float aw = fabsf(fmaf(-c, n.w, u.w) + rv.w);
        float a = wave_sum32(ax + ay + az + aw);

        if (lane == 0) out[e] = GAMMA_F - a;

        if (e2 >= E) break;
        e = e2; s = s2; d = d2; r = r2;
    }
}

// Kernel 1: normalized_norm[r] = norm_weight[r] / max(||norm_weight[r]||, eps)
// One wave per relation; each lane owns a float4 (32*4 = 128 dims).
__global__ __launch_bounds__(BLOCK)
void transh_normalize_kernel(const float* __restrict__ w,
                             float* __restrict__ out, int R) {
    const int lane = threadIdx.x & (WAVE - 1);
    const int r    = blockIdx.x * WPB + (threadIdx.x >> 5);
    if (r >= R) return;
    const float4* src = reinterpret_cast<const float4*>(w + (size_t)r * D);
    float4 v = src[lane];
    float ss = wave_sum32(v.x * v.x + v.y * v.y + v.z * v.z + v.w * v.w);
    float inv = 1.0f / fmaxf(sqrtf(ss), EPS_F);
    float4 o = make_float4(v.x * inv, v.y * inv, v.z * inv, v.w * inv);
    reinterpret_cast<float4*>(out + (size_t)r * D)[lane] = o;
}

extern "C" void kernel_launch(void* const* d_in, const int* in_sizes, int n_in,
                              void* d_out, int out_size, void* d_ws, size_t ws_size,
                              hipStream_t stream) {
    const float* node  = (const float*)d_in[0];   // [N,128]
    const float* normw = (const float*)d_in[1];   // [R,128]
    const float* relt  = (const float*)d_in[2];   // [R,128]
    const int*   src   = (const int*)d_in[3];     // [E]
    const int*   dst   = (const int*)d_in[4];     // [E]
    const int*   rid   = (const int*)d_in[5];     // [E]
    float*       out   = (float*)d_out;           // [E]

    const int E = in_sizes[3];
    const int R = in_sizes[1] / D;

    const size_t need = (size_t)R * D * sizeof(float);
    const bool prenorm = (ws_size >= need) && (d_ws != nullptr);
    const float* nrm = normw;

    if (prenorm) {
        float* wsn = (float*)d_ws;
        int nblk = (R + WPB - 1) / WPB;
        transh_normalize_kernel<<<nblk, BLOCK, 0, stream>>>(normw, wsn, R);
        nrm = wsn;
    }

    int blocks = 4096;                       // ~32K waves: plenty of MLP for L2 gathers
    int maxb = (E + WPB - 1) / WPB;
    if (blocks > maxb) blocks = maxb;
    transh_score_kernel<<<blocks, BLOCK, 0, stream>>>(
        node, nrm, relt, src, dst, rid, out, E, blocks * WPB, prenorm ? 1 : 0);
}